// ContrastLoss_11605001634273
// MI455X (gfx1250) — compile-verified
//
#include <hip/hip_runtime.h>
#include <hip/hip_bf16.h>
#include <math.h>

typedef __attribute__((ext_vector_type(2))) float v2f;
typedef __attribute__((ext_vector_type(8))) float v8f;

#define C_FEAT   256
#define HW       65536        // H*W
#define N_CLS    19
#define KM_      10
#define P_TOT    190          // 19*10 prototypes
#define P_PAD    192          // 12 tiles of 16
#define KC       64           // channels per K-chunk
#define LDA      68           // padded LDS row stride (floats): 68%64==4 -> conflict-free
#define LDB      68
#define LDD      204
#define TILE_M   16
#define INV_T    10.0f        // 1/TEMPERATURE

__global__ __launch_bounds__(128)
void contrast_gemm_kernel(const float* __restrict__ embs,
                          const float* __restrict__ proto,
                          const int*   __restrict__ anchor_idx,
                          const int*   __restrict__ anchor_labels,
                          const unsigned char* __restrict__ proto_mask,
                          float* __restrict__ row_out,
                          int Mtot)
{
    __shared__ float smem[TILE_M * LDA + P_PAD * LDB]; // 56,576 B
    float* As = smem;
    float* Bs = smem + TILE_M * LDA;
    float* Ds = smem; // aliased after GEMM phase (needs 16*204 floats)

    const int tid   = threadIdx.x;
    const int lane  = tid & 31;
    const int wave  = tid >> 5;     // 0..3, owns N-tiles wave*3 .. wave*3+2
    const int lm    = lane & 15;
    const int hs    = lane >> 4;    // half-wave select
    const int mbase = blockIdx.x * TILE_M;

    v8f acc[3];
    acc[0] = (v8f){}; acc[1] = (v8f){}; acc[2] = (v8f){};

    for (int kc = 0; kc < C_FEAT; kc += KC) {
        __syncthreads();
        // ---- gather A tile: 16 anchors x 64 channels (strided HBM gather) ----
        for (int i = tid; i < TILE_M * KC; i += 128) {
            int m   = i >> 6;
            int c   = i & (KC - 1);
            int gm  = mbase + m;
            int f   = (gm < Mtot) ? anchor_idx[gm] : 0;
            int bb  = f >> 16;          // f / (H*W)
            int pix = f & (HW - 1);     // f % (H*W)
            As[m * LDA + c] =
                embs[(size_t)(bb * C_FEAT + kc + c) * HW + (size_t)pix];
        }
        // ---- load B tile: 192 protos x 64 channels, float4, zero-pad rows ----
        for (int i = tid; i < P_PAD * (KC / 4); i += 128) {
            int p  = i >> 4;
            int c4 = (i & 15) << 2;
            float4 v = make_float4(0.f, 0.f, 0.f, 0.f);
            if (p < P_TOT)
                v = *reinterpret_cast<const float4*>(
                        proto + (size_t)p * C_FEAT + kc + c4);
            Bs[p * LDB + c4 + 0] = v.x;
            Bs[p * LDB + c4 + 1] = v.y;
            Bs[p * LDB + c4 + 2] = v.z;
            Bs[p * LDB + c4 + 3] = v.w;
        }
        __syncthreads();
        // ---- 16 k-steps of full-precision V_WMMA_F32_16X16X4_F32 ----
        for (int ks = 0; ks < KC; ks += 4) {
            int kk = ks + hs * 2;
            v2f a = *reinterpret_cast<const v2f*>(&As[lm * LDA + kk]);
#pragma unroll
            for (int t = 0; t < 3; ++t) {
                int n = (wave * 3 + t) * 16 + lm;
                v2f bfr = *reinterpret_cast<const v2f*>(&Bs[n * LDB + kk]);
                acc[t] = __builtin_amdgcn_wmma_f32_16x16x4_f32(
                    false, a, false, bfr, (short)0, acc[t], false, false);
            }
        }
    }
    __syncthreads();
    // ---- spill D tiles to LDS, scaled by 1/TEMPERATURE ----
#pragma unroll
    for (int t = 0; t < 3; ++t) {
        int col = (wave * 3 + t) * 16 + lm;
#pragma unroll
        for (int v = 0; v < 8; ++v) {
            int row = v + 8 * hs;
            Ds[row * LDD + col] = acc[t][v] * INV_T;
        }
    }
    __syncthreads();
    // ---- per-row masked softmax/log-prob reduction (one thread per row) ----
    if (tid < TILE_M) {
        int gm = mbase + tid;
        if (gm < Mtot) {
            int lbl = anchor_labels[gm];
            const float* drow = &Ds[tid * LDD];
            // pass 1: max over valid prototypes
            float mx = -INFINITY;
            for (int p = 0; p < P_TOT; ++p)
                if (proto_mask[p]) mx = fmaxf(mx, drow[p]);
            if (mx == -INFINITY) mx = 0.f;
            // pass 2: sum of exp over negatives (valid && label mismatch)
            float negsum = 0.f;
            for (int p = 0; p < P_TOT; ++p)
                if (proto_mask[p] && (p / KM_ != lbl))
                    negsum += expf(drow[p] - mx);
            // pass 3: positive log-prob terms
            float lpsum = 0.f;
            float cnt = 0.f;
            for (int p = 0; p < P_TOT; ++p)
                if (proto_mask[p] && (p / KM_ == lbl)) {
                    float lg = drow[p] - mx;
                    lpsum += lg - logf(expf(lg) + negsum + 1e-12f);
                    cnt += 1.f;
                }
            row_out[gm] = lpsum / (cnt + 1e-12f);
        }
    }
}

__global__ __launch_bounds__(256)
void contrast_reduce_kernel(const float* __restrict__ row_out,
                            float* __restrict__ out, int Mtot)
{
    __shared__ float s[256];
    float sum = 0.f;
    for (int i = threadIdx.x; i < Mtot; i += 256) sum += row_out[i];
    s[threadIdx.x] = sum;
    __syncthreads();
    for (int o = 128; o > 0; o >>= 1) {
        if ((int)threadIdx.x < o) s[threadIdx.x] += s[threadIdx.x + o];
        __syncthreads();
    }
    if (threadIdx.x == 0) out[0] = -s[0] / (float)Mtot;
}

extern "C" void kernel_launch(void* const* d_in, const int* in_sizes, int n_in,
                              void* d_out, int out_size, void* d_ws, size_t ws_size,
                              hipStream_t stream) {
    const float*         embs  = (const float*)d_in[0];
    const float*         proto = (const float*)d_in[1];
    const int*           aidx  = (const int*)d_in[2];
    const int*           albl  = (const int*)d_in[3];
    const unsigned char* pmask = (const unsigned char*)d_in[4];
    float* out = (float*)d_out;
    float* ws  = (float*)d_ws;   // per-row mean_log_prob_pos, M floats

    const int M = in_sizes[2];
    const int nblocks = (M + TILE_M - 1) / TILE_M;

    contrast_gemm_kernel<<<nblocks, 128, 0, stream>>>(
        embs, proto, aidx, albl, pmask, ws, M);
    contrast_reduce_kernel<<<1, 256, 0, stream>>>(ws, out, M);
}